// PointProcessGenerator_72335839199666
// MI455X (gfx1250) — compile-verified
//
#include <hip/hip_runtime.h>
#include <math.h>

// ---------------------------------------------------------------------------
// PointProcessGenerator on MI455X (gfx1250, wave32)
//   Phase 1: sequential LSTM rollout, 1 workgroup, V_WMMA_F32_16X16X4_F32
//   Phase 2: MMD Gaussian-kernel sum, folded into one signed symmetric sum
// ---------------------------------------------------------------------------

typedef __attribute__((ext_vector_type(2))) float v2f;
typedef __attribute__((ext_vector_type(8))) float v8f;

#define HIDDEN 64
#define SEQLEN 128
#define BATCH  64
#define HXPAD  66   // padded hx row stride (floats) -> conflict-free A-frag loads

__device__ __forceinline__ float sigmoidf_(float x) {
    return 1.0f / (1.0f + __expf(-x));
}

// ---------------------------------------------------------------------------
// Phase 1: one workgroup (512 threads = 16 waves), W_hh resident in VGPRs.
// Wave w: Mbase = (w&3)*16 (batch tile), Hbase = (w>>2)*16 (hidden tile).
// Each wave computes gate tiles i/f/g/o at columns Hbase+{0,64,128,192}
// so the full LSTM nonlinearity happens on its own D fragments.
// ---------------------------------------------------------------------------
__global__ __launch_bounds__(512)
void pp_rollout_kernel(const float* __restrict__ u,     // [64][128] uniform pool
                       const float* __restrict__ W_ih,  // [256] (4H x 1)
                       const float* __restrict__ W_hh,  // [256][64]
                       const float* __restrict__ b_ih,  // [256]
                       const float* __restrict__ b_hh,  // [256]
                       const float* __restrict__ V_w,   // [64]
                       const float* __restrict__ V_b,   // [1]
                       float* __restrict__ lt)          // out: [64][128]
{
    __shared__ float hx[BATCH * HXPAD];
    __shared__ float cum[BATCH];
    __shared__ float vw[HIDDEN];

    const int tid   = threadIdx.x;
    const int lane  = tid & 31;
    const int wave  = tid >> 5;
    const int Mbase = (wave & 3) * 16;
    const int Hbase = (wave >> 2) * 16;
    const int lrow  = lane & 15;
    const int khalf = lane >> 4;      // 0: lanes 0-15 hold K={0,1}; 1: K={2,3}
    const int koff  = khalf * 2;
    const int hcol  = Hbase + lrow;

    // ---- init state ----
    for (int i = tid; i < BATCH * HXPAD; i += 512) hx[i] = 0.0f;
    if (tid < BATCH)  cum[tid] = 0.0f;
    if (tid < HIDDEN) vw[tid]  = V_w[tid];

    // ---- preload W_hh B-fragments (loop-invariant) into registers ----
    // B[k][n] = W_hh[g*64 + Hbase + n][k]; f32 B-frag layout (K=4):
    // lanes 0-15 hold (K=k0,k0+1) of column n=lane; lanes 16-31 hold (k0+2,k0+3)
    v2f wb[4][16];
#pragma unroll
    for (int g = 0; g < 4; ++g) {
        const float* wr = W_hh + (g * HIDDEN + Hbase + lrow) * HIDDEN + koff;
#pragma unroll
        for (int kk = 0; kk < 16; ++kk) {
            wb[g][kk][0] = wr[kk * 4 + 0];
            wb[g][kk][1] = wr[kk * 4 + 1];
        }
    }

    // per-lane gate constants (depend only on column hcol)
    float wih_r[4], bias_r[4];
#pragma unroll
    for (int g = 0; g < 4; ++g) {
        wih_r[g]  = W_ih[g * HIDDEN + hcol];
        bias_r[g] = b_ih[g * HIDDEN + hcol] + b_hh[g * HIDDEN + hcol];
    }
    const float vb = V_b[0];

    // cx in registers, laid out exactly like a 16x16 f32 D fragment
    float cx[8];
#pragma unroll
    for (int r = 0; r < 8; ++r) cx[r] = 0.0f;

    __syncthreads();

    for (int s = 0; s < SEQLEN; ++s) {
        // ---- phase A: sigma / cum update (threads 0..63) ----
        if (tid < BATCH) {
            float z = vb;
#pragma unroll 8
            for (int k = 0; k < HIDDEN; ++k) z += hx[tid * HXPAD + k] * vw[k];
            float sigma = (z > 0.0f) ? (z + 1.0f) : __expf(z);   // elu(z)+1
            float c = cum[tid] + (-__logf(u[tid * SEQLEN + s])) / sigma;
            cum[tid] = c;
            lt[tid * SEQLEN + s] = c;                            // emit cum AFTER update
        }

        // ---- gates = hx @ W_hh^T via f32 WMMA (K=4 steps, fully unrolled) ----
        v8f acc0 = {}, acc1 = {}, acc2 = {}, acc3 = {};
#pragma unroll
        for (int kk = 0; kk < 16; ++kk) {
            const int k0 = kk * 4 + koff;
            v2f a;
            a[0] = hx[(Mbase + lrow) * HXPAD + k0];
            a[1] = hx[(Mbase + lrow) * HXPAD + k0 + 1];
            acc0 = __builtin_amdgcn_wmma_f32_16x16x4_f32(false, a, false, wb[0][kk], (short)0, acc0, false, false);
            acc1 = __builtin_amdgcn_wmma_f32_16x16x4_f32(false, a, false, wb[1][kk], (short)0, acc1, false, false);
            acc2 = __builtin_amdgcn_wmma_f32_16x16x4_f32(false, a, false, wb[2][kk], (short)0, acc2, false, false);
            acc3 = __builtin_amdgcn_wmma_f32_16x16x4_f32(false, a, false, wb[3][kk], (short)0, acc3, false, false);
        }

        __syncthreads();   // hx reads + cum writes complete

        // ---- phase B: LSTM nonlinearity on D fragments, write hx ----
        // D layout: VGPR r, lanes 0-15 -> row Mbase+r, lanes 16-31 -> row Mbase+r+8
#pragma unroll
        for (int r = 0; r < 8; ++r) {
            const int brow = Mbase + r + khalf * 8;
            const float c  = cum[brow];
            float iv = sigmoidf_(acc0[r] + c * wih_r[0] + bias_r[0]);
            float fv = sigmoidf_(acc1[r] + c * wih_r[1] + bias_r[1]);
            float gv = tanhf    (acc2[r] + c * wih_r[2] + bias_r[2]);
            float ov = sigmoidf_(acc3[r] + c * wih_r[3] + bias_r[3]);
            float cxn = fv * cx[r] + iv * gv;
            cx[r] = cxn;
            hx[brow * HXPAD + hcol] = ov * tanhf(cxn);
        }
        __syncthreads();   // hx stable before next step's reads
    }
}

// ---------------------------------------------------------------------------
// Phase 2: loss = sum_{i,j} s_i s_j exp(-(x_i-x_j)^2),
//   x = [expert(8192); learner(8192)], s = +mask / -mask  (BW = 1)
// 512 blocks = 64 i-chunks x 8 j-chunks; deterministic tree reduction.
// ---------------------------------------------------------------------------
__device__ __forceinline__ void load_point(const float* __restrict__ expert,
                                           const float* __restrict__ lt,
                                           int idx, float& x, float& s) {
    if (idx < 8192) {
        x = expert[idx];
        s = ((x > 0.0f) && (x < 20.0f)) ? 1.0f : 0.0f;
    } else {
        x = lt[idx - 8192];
        s = ((x > 0.0f) && (x < 20.0f)) ? -1.0f : 0.0f;
    }
}

__global__ __launch_bounds__(256)
void mmd_partial_kernel(const float* __restrict__ expert,  // [8192]
                        const float* __restrict__ lt,      // [8192]
                        float* __restrict__ partial)       // [512]
{
    __shared__ float xs[256];
    __shared__ float ss[256];
    __shared__ float red[256];

    const int tid = threadIdx.x;
    const int ci  = blockIdx.x & 63;   // i-chunk (64 x 256 = 16384)
    const int cj  = blockIdx.x >> 6;   // j-chunk (8 x 2048 = 16384)
    const int i   = ci * 256 + tid;

    float xi, si;
    load_point(expert, lt, i, xi, si);

    float acc = 0.0f;
    for (int jt = cj * 2048; jt < (cj + 1) * 2048; jt += 256) {
        float xj, sj;
        load_point(expert, lt, jt + tid, xj, sj);
        __syncthreads();
        xs[tid] = xj;
        ss[tid] = sj;
        __syncthreads();
#pragma unroll 8
        for (int k = 0; k < 256; ++k) {
            float d = xi - xs[k];
            acc += ss[k] * __expf(-d * d);
        }
    }
    acc *= si;

    red[tid] = acc;
    __syncthreads();
    for (int off = 128; off > 0; off >>= 1) {
        if (tid < off) red[tid] += red[tid + off];
        __syncthreads();
    }
    if (tid == 0) partial[blockIdx.x] = red[0];
}

__global__ __launch_bounds__(512)
void mmd_reduce_kernel(const float* __restrict__ partial, float* __restrict__ out) {
    __shared__ float red[512];
    const int tid = threadIdx.x;
    red[tid] = partial[tid];
    __syncthreads();
    for (int off = 256; off > 0; off >>= 1) {
        if (tid < off) red[tid] += red[tid + off];
        __syncthreads();
    }
    if (tid == 0) out[0] = red[0];
}

// ---------------------------------------------------------------------------
extern "C" void kernel_launch(void* const* d_in, const int* in_sizes, int n_in,
                              void* d_out, int out_size, void* d_ws, size_t ws_size,
                              hipStream_t stream) {
    (void)in_sizes; (void)n_in; (void)out_size; (void)ws_size;
    const float* u   = (const float*)d_in[0];  // rand_uniform_pool [64][128]
    const float* et  = (const float*)d_in[1];  // expert_time       [64][128]
    const float* Wih = (const float*)d_in[2];  // [256][1]
    const float* Whh = (const float*)d_in[3];  // [256][64]
    const float* bih = (const float*)d_in[4];  // [256]
    const float* bhh = (const float*)d_in[5];  // [256]
    const float* Vw  = (const float*)d_in[6];  // [1][64]
    const float* Vb  = (const float*)d_in[7];  // [1]

    float* lt      = (float*)d_ws;     // [8192] learner times
    float* partial = lt + 8192;        // [512]  block partial sums

    pp_rollout_kernel<<<1, 512, 0, stream>>>(u, Wih, Whh, bih, bhh, Vw, Vb, lt);
    mmd_partial_kernel<<<512, 256, 0, stream>>>(et, lt, partial);
    mmd_reduce_kernel<<<1, 512, 0, stream>>>(partial, (float*)d_out);
}